// LinkPredictor_26225070310150
// MI455X (gfx1250) — compile-verified
//
#include <hip/hip_runtime.h>

#define N_P 200000
#define N_L 20000
#define EMB 64
#define E_PP 2000000
#define E_PL 1000000
#define E_LP 1000000

typedef __attribute__((ext_vector_type(2))) float v2f;
typedef __attribute__((ext_vector_type(8))) float v8f;

// ---------------------------------------------------------------- utilities

__global__ void fill_zero4(float4* __restrict__ p, long n4) {
    long stride = (long)gridDim.x * blockDim.x;
    for (long i = (long)blockIdx.x * blockDim.x + threadIdx.x; i < n4; i += stride)
        p[i] = make_float4(0.f, 0.f, 0.f, 0.f);
}

__global__ void add_kernel(const float* __restrict__ a, const float* __restrict__ b,
                           float* __restrict__ o, int n) {
    int i = blockIdx.x * blockDim.x + threadIdx.x;
    if (i < n) o[i] = a[i] + b[i];
}

// ---------------------------------------------------------------- CSR build

__global__ void count_int_kernel(const int* __restrict__ dst, int* __restrict__ cnt, int E) {
    long stride = (long)gridDim.x * blockDim.x;
    for (long i = (long)blockIdx.x * blockDim.x + threadIdx.x; i < E; i += stride)
        atomicAdd(&cnt[dst[i]], 1);
}

// exclusive scan: rowptr[0]=0, rowptr[i+1]=sum(cnt[0..i]); single block, wave32 shuffles
__global__ __launch_bounds__(1024) void scan_kernel(const int* __restrict__ cnt,
                                                    int* __restrict__ rowptr, int n) {
    __shared__ int wsum[32];
    __shared__ int carry;
    const int tid = threadIdx.x;
    const int lane = tid & 31;
    const int wid = tid >> 5;
    if (tid == 0) carry = 0;
    __syncthreads();
    for (int base = 0; base < n; base += 1024) {
        int i = base + tid;
        int x = (i < n) ? cnt[i] : 0;
#pragma unroll
        for (int off = 1; off < 32; off <<= 1) {       // inclusive wave scan
            int y = __shfl_up(x, off, 32);
            if (lane >= off) x += y;
        }
        if (lane == 31) wsum[wid] = x;
        __syncthreads();
        if (wid == 0) {                                 // scan the 32 wave totals
            int s = wsum[lane];
#pragma unroll
            for (int off = 1; off < 32; off <<= 1) {
                int y = __shfl_up(s, off, 32);
                if (lane >= off) s += y;
            }
            wsum[lane] = s;
        }
        __syncthreads();
        int incl = x + ((wid > 0) ? wsum[wid - 1] : 0);
        int c = carry;
        if (i < n) rowptr[i + 1] = c + incl;
        __syncthreads();
        if (tid == 1023) carry = c + incl;              // tile total
        __syncthreads();
    }
    if (threadIdx.x == 0) rowptr[0] = 0;
}

__global__ void fill_csr_kernel(const int* __restrict__ src, const int* __restrict__ dst,
                                const int* __restrict__ rowptr, int* __restrict__ cursor,
                                int* __restrict__ col, int E) {
    long stride = (long)gridDim.x * blockDim.x;
    for (long i = (long)blockIdx.x * blockDim.x + threadIdx.x; i < E; i += stride) {
        int d = dst[i];
        int pos = atomicAdd(&cursor[d], 1);
        col[rowptr[d] + pos] = src[i];
    }
}

// deterministic neighbor order + better gather locality: sort each bucket by src id
__global__ void sort_buckets_kernel(const int* __restrict__ rowptr, int* __restrict__ col, int n) {
    int node = blockIdx.x * blockDim.x + threadIdx.x;
    if (node >= n) return;
    int b = rowptr[node], e = rowptr[node + 1];
    for (int i = b + 1; i < e; ++i) {
        int v = col[i];
        int j = i - 1;
        while (j >= b && col[j] > v) { col[j + 1] = col[j]; --j; }
        col[j + 1] = v;
    }
}

// ---------------------------------------------------------------- gather-reduce
// agg[node,:] = sum over CSR neighbors of x[src,:]; one block per dst node,
// thread t owns feature t (coalesced 512B row loads, no atomics, writes all rows).
template <int D>
__global__ __launch_bounds__(128) void gather_kernel(const float* __restrict__ x,
                                                     const int* __restrict__ rowptr,
                                                     const int* __restrict__ col,
                                                     float* __restrict__ agg) {
    const int node = blockIdx.x;
    const int t = threadIdx.x;        // blockDim.x == D
    int b = rowptr[node], e = rowptr[node + 1];
    float acc0 = 0.f, acc1 = 0.f;
    int j = b;
    for (; j + 1 < e; j += 2) {
        int s0 = col[j], s1 = col[j + 1];
        acc0 += x[(size_t)s0 * D + t];
        acc1 += x[(size_t)s1 * D + t];
    }
    if (j < e) acc0 += x[(size_t)col[j] * D + t];
    agg[(size_t)node * D + t] = acc0 + acc1;
}

// ---------------------------------------------------------------- fused WMMA GEMM
// out[row0:row0+16, :N] = sum_t (A_t * rowscale_t) @ W_t + bias0 + bias1 (optional relu)
// rowscale_t = 1/max(deg,1) from CSR rowptr when rp_t != null (SAGE mean), else 1.
// Each wave owns a 16x16 tile; K stepped by 4 with V_WMMA_F32_16X16X4_F32 (exact fp32).
__global__ __launch_bounds__(256) void gemm3(
    const float* __restrict__ A0, const float* __restrict__ W0, const int* __restrict__ rp0, int K0,
    const float* __restrict__ A1, const float* __restrict__ W1, const int* __restrict__ rp1, int K1,
    const float* __restrict__ A2, const float* __restrict__ W2, const int* __restrict__ rp2, int K2,
    const float* __restrict__ bias0, const float* __restrict__ bias1,
    float* __restrict__ out, int N, int relu)
{
    __shared__ float sA[16 * 130];   // pad stride 130 -> conflict-free b64 reads
    __shared__ float sSc[16];

    const int row0 = blockIdx.x * 16;
    const int wave = threadIdx.x >> 5;
    const int lane = threadIdx.x & 31;
    const int col0 = wave * 16;
    const int t  = lane & 15;   // row (A) / col (B,C) within tile
    const int hf = lane >> 4;   // K-pair / M-half selector

    v8f acc = {};

    const float* As[3] = {A0, A1, A2};
    const float* Ws[3] = {W0, W1, W2};
    const int*   Rs[3] = {rp0, rp1, rp2};
    const int    Ks[3] = {K0, K1, K2};

    for (int term = 0; term < 3; ++term) {
        const float* A = As[term];
        if (A == nullptr) continue;          // uniform branch
        const int K = Ks[term];              // 64 or 128 (power of two)
        const int* rp = Rs[term];

        __syncthreads();                     // protect previous term's LDS
        if (threadIdx.x < 16) {
            float s = 1.0f;
            if (rp) {
                int r = row0 + (int)threadIdx.x;
                int deg = rp[r + 1] - rp[r];
                s = 1.0f / (float)(deg > 1 ? deg : 1);
            }
            sSc[threadIdx.x] = s;
        }
        __syncthreads();

        const int ksh = __builtin_ctz(K);
        for (int i = threadIdx.x; i < (K << 4); i += blockDim.x) {
            int r = i >> ksh;
            int c = i & (K - 1);
            sA[r * 130 + c] = A[(size_t)(row0 + r) * K + c] * sSc[r];
        }
        __syncthreads();

        const float* W = Ws[term];
        for (int k0 = 0; k0 < K; k0 += 4) {
            int kk = k0 + 2 * hf;
            v2f a, b;
            // A 16x4 frag: lanes 0-15 -> K={k0,k0+1}, lanes 16-31 -> K={k0+2,k0+3}
            a.x = sA[t * 130 + kk];
            a.y = sA[t * 130 + kk + 1];
            // B 4x16 frag (mirrored): lane t -> column col0+t
            b.x = W[(size_t)kk * N + col0 + t];
            b.y = W[(size_t)(kk + 1) * N + col0 + t];
            acc = __builtin_amdgcn_wmma_f32_16x16x4_f32(
                false, a, false, b, (short)0, acc, false, false);
        }
    }
    __syncthreads();

    float bb = 0.0f;
    if (bias0) bb += bias0[col0 + t];
    if (bias1) bb += bias1[col0 + t];
#pragma unroll
    for (int i = 0; i < 8; ++i) {
        // C/D layout: VGPR i, lanes 0-15 -> M=i ; lanes 16-31 -> M=8+i ; N=lane&15
        float v = acc[i] + bb;
        if (relu) v = fmaxf(v, 0.0f);
        out[(size_t)(row0 + i + 8 * hf) * N + col0 + t] = v;
    }
}

// ---------------------------------------------------------------- launch

static inline unsigned gsblocks(long total, int per) {
    long b = (total + per - 1) / per;
    if (b > (1L << 20)) b = 1L << 20;
    if (b < 1) b = 1;
    return (unsigned)b;
}

extern "C" void kernel_launch(void* const* d_in, const int* in_sizes, int n_in,
                              void* d_out, int out_size, void* d_ws, size_t ws_size,
                              hipStream_t stream) {
    const float* x_paper = (const float*)d_in[0];
    const float* x_label = (const float*)d_in[1];
    const int*   ei_pp   = (const int*)d_in[2];
    const int*   ei_pl   = (const int*)d_in[3];
    const int*   ei_lp   = (const int*)d_in[4];
    const float* Wl1_pp = (const float*)d_in[5];
    const float* bl1_pp = (const float*)d_in[6];
    const float* Wr1_pp = (const float*)d_in[7];
    const float* Wl1_pl = (const float*)d_in[8];
    const float* bl1_pl = (const float*)d_in[9];
    const float* Wr1_pl = (const float*)d_in[10];
    const float* Wl1_lp = (const float*)d_in[11];
    const float* bl1_lp = (const float*)d_in[12];
    const float* Wr1_lp = (const float*)d_in[13];
    const float* Wl2_pp = (const float*)d_in[14];
    const float* bl2_pp = (const float*)d_in[15];
    const float* Wr2_pp = (const float*)d_in[16];
    const float* Wl2_pl = (const float*)d_in[17];
    const float* bl2_pl = (const float*)d_in[18];
    const float* Wr2_pl = (const float*)d_in[19];
    const float* Wl2_lp = (const float*)d_in[20];
    const float* bl2_lp = (const float*)d_in[21];
    const float* Wr2_lp = (const float*)d_in[22];

    // ---- workspace layout ----
    float* ws   = (float*)d_ws;
    float* aggA = ws;                          // [N_P,128]  agg_pp  / agg2_pp
    float* aggB = aggA + (size_t)N_P * 128;    // [N_P,128]  agg_lp (64 cols L1) / agg2_lp
    float* aggC = aggB + (size_t)N_P * 128;    // [N_L,128]  agg_pl  / agg2_pl
    float* h_p  = aggC + (size_t)N_L * 128;    // [N_P,128]
    float* h_l  = h_p  + (size_t)N_P * 128;    // [N_L,128]
    float* Wc1  = h_l  + (size_t)N_L * 128;    // [128,128] = Wr1_pp + Wr1_lp
    float* Wc2  = Wc1 + 128 * 128;             // [128,64]  = Wr2_pp + Wr2_lp
    int* rp_pp  = (int*)(Wc2 + 128 * 64);      // [N_P+1]
    int* rp_lp  = rp_pp + (N_P + 1);           // [N_P+1]
    int* rp_pl  = rp_lp + (N_P + 1);           // [N_L+1]
    int* cnt_pp = rp_pl + (N_L + 1);           // [N_P] counts, then cursors
    int* cnt_lp = cnt_pp + N_P;                // [N_P]
    int* cnt_pl = cnt_lp + N_P;                // [N_L]
    int* col_pp = cnt_pl + N_L;                // [E_PP]
    int* col_lp = col_pp + E_PP;               // [E_LP]
    int* col_pl = col_lp + E_LP;               // [E_PL]

    const int* src_pp = ei_pp; const int* dst_pp = ei_pp + E_PP;
    const int* src_pl = ei_pl; const int* dst_pl = ei_pl + E_PL;
    const int* src_lp = ei_lp; const int* dst_lp = ei_lp + E_LP;

    const long NCNT = (long)N_P + N_P + N_L;   // counts region is contiguous

    // ---- CSR build (edge lists identical for both layers; amortized over 6 aggs) ----
    fill_zero4<<<gsblocks(NCNT / 4, 256), 256, 0, stream>>>((float4*)cnt_pp, NCNT / 4);
    count_int_kernel<<<gsblocks(E_PP, 256), 256, 0, stream>>>(dst_pp, cnt_pp, E_PP);
    count_int_kernel<<<gsblocks(E_LP, 256), 256, 0, stream>>>(dst_lp, cnt_lp, E_LP);
    count_int_kernel<<<gsblocks(E_PL, 256), 256, 0, stream>>>(dst_pl, cnt_pl, E_PL);
    scan_kernel<<<1, 1024, 0, stream>>>(cnt_pp, rp_pp, N_P);
    scan_kernel<<<1, 1024, 0, stream>>>(cnt_lp, rp_lp, N_P);
    scan_kernel<<<1, 1024, 0, stream>>>(cnt_pl, rp_pl, N_L);
    fill_zero4<<<gsblocks(NCNT / 4, 256), 256, 0, stream>>>((float4*)cnt_pp, NCNT / 4); // cursors
    fill_csr_kernel<<<gsblocks(E_PP, 256), 256, 0, stream>>>(src_pp, dst_pp, rp_pp, cnt_pp, col_pp, E_PP);
    fill_csr_kernel<<<gsblocks(E_LP, 256), 256, 0, stream>>>(src_lp, dst_lp, rp_lp, cnt_lp, col_lp, E_LP);
    fill_csr_kernel<<<gsblocks(E_PL, 256), 256, 0, stream>>>(src_pl, dst_pl, rp_pl, cnt_pl, col_pl, E_PL);
    sort_buckets_kernel<<<gsblocks(N_P, 256), 256, 0, stream>>>(rp_pp, col_pp, N_P);
    sort_buckets_kernel<<<gsblocks(N_P, 256), 256, 0, stream>>>(rp_lp, col_lp, N_P);
    sort_buckets_kernel<<<gsblocks(N_L, 256), 256, 0, stream>>>(rp_pl, col_pl, N_L);

    // ---- combined root weights ----
    add_kernel<<<(128 * 128 + 255) / 256, 256, 0, stream>>>(Wr1_pp, Wr1_lp, Wc1, 128 * 128);
    add_kernel<<<(128 * 64 + 255) / 256, 256, 0, stream>>>(Wr2_pp, Wr2_lp, Wc2, 128 * 64);

    // ---- layer 1: gather-reduce (no atomics, fully coalesced) ----
    gather_kernel<128><<<N_P, 128, 0, stream>>>(x_paper, rp_pp, col_pp, aggA);
    gather_kernel< 64><<<N_P,  64, 0, stream>>>(x_label, rp_lp, col_lp, aggB);
    gather_kernel<128><<<N_L, 128, 0, stream>>>(x_paper, rp_pl, col_pl, aggC);

    // h_p = relu(mean_pp@Wl1_pp + mean_lp@Wl1_lp + x_paper@(Wr1_pp+Wr1_lp) + bl1_pp + bl1_lp)
    gemm3<<<N_P / 16, 256, 0, stream>>>(aggA, Wl1_pp, rp_pp, 128,
                                        aggB, Wl1_lp, rp_lp, 64,
                                        x_paper, Wc1, nullptr, 128,
                                        bl1_pp, bl1_lp, h_p, 128, 1);
    // h_l = relu(mean_pl@Wl1_pl + x_label@Wr1_pl + bl1_pl)
    gemm3<<<N_L / 16, 256, 0, stream>>>(aggC, Wl1_pl, rp_pl, 128,
                                        x_label, Wr1_pl, nullptr, 64,
                                        nullptr, nullptr, nullptr, 0,
                                        bl1_pl, nullptr, h_l, 128, 1);

    // ---- layer 2: gather-reduce over hidden features ----
    gather_kernel<128><<<N_P, 128, 0, stream>>>(h_p, rp_pp, col_pp, aggA);
    gather_kernel<128><<<N_P, 128, 0, stream>>>(h_l, rp_lp, col_lp, aggB);
    gather_kernel<128><<<N_L, 128, 0, stream>>>(h_p, rp_pl, col_pl, aggC);

    // ---- layer 2 GEMMs -> d_out ----
    float* z_p = (float*)d_out;
    float* z_l = z_p + (size_t)N_P * EMB;
    // z_p = mean2_pp@Wl2_pp + mean2_lp@Wl2_lp + h_p@(Wr2_pp+Wr2_lp) + bl2_pp + bl2_lp
    gemm3<<<N_P / 16, 128, 0, stream>>>(aggA, Wl2_pp, rp_pp, 128,
                                        aggB, Wl2_lp, rp_lp, 128,
                                        h_p, Wc2, nullptr, 128,
                                        bl2_pp, bl2_lp, z_p, 64, 0);
    // z_l = mean2_pl@Wl2_pl + h_l@Wr2_pl + bl2_pl
    gemm3<<<N_L / 16, 128, 0, stream>>>(aggC, Wl2_pl, rp_pl, 128,
                                        h_l, Wr2_pl, nullptr, 128,
                                        nullptr, nullptr, nullptr, 0,
                                        bl2_pl, nullptr, z_l, 64, 0);
}